// LM_MLE_70171175682103
// MI455X (gfx1250) — compile-verified
//
#include <hip/hip_runtime.h>

typedef __attribute__((ext_vector_type(2))) float v2f;
typedef __attribute__((ext_vector_type(4))) float v4f;
typedef __attribute__((ext_vector_type(8))) float v8f;

#define LAMBDA_ 0.01f
#define KEPS_F  2.220446049250313e-13f   /* 1000 * f64 eps */
#define WAVES_PER_BLOCK 4
#define LDS_FLOATS_PER_WAVE 2112         /* 2048 Xrow + 64 scratch */

__global__ __launch_bounds__(128) void lm_mle_wmma_kernel(
    const float* __restrict__ smp,   // [B,16,16]
    const float* __restrict__ mu,    // [B,16,16]
    const float* __restrict__ jac,   // [B,16,16,5]
    const float* __restrict__ cur,   // [B,5]
    const float* __restrict__ prm,   // [5,2] (min,max)
    float* __restrict__ out,         // [B,5]
    int B)
{
    __shared__ float lds[WAVES_PER_BLOCK * LDS_FLOATS_PER_WAVE];

    const int lane = threadIdx.x & 31;
    const int wave = threadIdx.x >> 5;
    const int b    = blockIdx.x * WAVES_PER_BLOCK + wave;
    if (b >= B) return;

    float* Xrow = lds + wave * LDS_FLOATS_PER_WAVE;  // [256][8]: s*j0..s*j4, t, 0, 0
    float* sol  = Xrow + 2048;                       // 6x6 corner of Gram matrix

    const float* smpB = smp + (size_t)b * 256;
    const float* muB  = mu  + (size_t)b * 256;
    const float* jacB = jac + (size_t)b * 1280;

    // ---------------- Phase 1: build scaled design rows in LDS ----------------
    // Each lane owns pixels k0..k0+7.
    const int k0 = lane * 8;

    float sva[8], mva[8];
    *(v4f*)&sva[0] = __builtin_nontemporal_load((const v4f*)(smpB + k0));
    *(v4f*)&sva[4] = __builtin_nontemporal_load((const v4f*)(smpB + k0 + 4));
    *(v4f*)&mva[0] = __builtin_nontemporal_load((const v4f*)(muB  + k0));
    *(v4f*)&mva[4] = __builtin_nontemporal_load((const v4f*)(muB  + k0 + 4));

    float jf[40];                    // 8 pixels x 5 jac entries, 160B aligned chunk
    #pragma unroll
    for (int i = 0; i < 10; ++i)
        *(v4f*)&jf[i * 4] = __builtin_nontemporal_load((const v4f*)(jacB + k0 * 5 + i * 4));

    #pragma unroll
    for (int i = 0; i < 8; ++i) {
        const float sv = sva[i];
        const float m  = mva[i];
        // mupos = (mu>0)*mu + (mu<0)*KEPS  (exact reference semantics)
        const float mupos = (m > 0.0f) ? m : ((m < 0.0f) ? KEPS_F : 0.0f);
        const float sq = __builtin_sqrtf(sv);
        const float s  = sq / mupos;              // sqrt(dweight)
        const float t  = (sv - mupos) / sq;       // weight / s
        v4f r0, r1;
        r0.x = s * jf[i * 5 + 0];
        r0.y = s * jf[i * 5 + 1];
        r0.z = s * jf[i * 5 + 2];
        r0.w = s * jf[i * 5 + 3];
        r1.x = s * jf[i * 5 + 4];
        r1.y = t;
        r1.z = 0.0f;
        r1.w = 0.0f;
        *(v4f*)(Xrow + (k0 + i) * 8)     = r0;
        *(v4f*)(Xrow + (k0 + i) * 8 + 4) = r1;
    }
    // Same-wave LDS ops are in-order (DScnt): no barrier needed; LDS is wave-private.

    // ---------------- Phase 2: Gram accumulation via WMMA f32 16x16x4 --------
    // Operand layout (lane l, c = l&15, h = l>>4, VGPR v):
    //   A(m=c, k=2h+v) and B(k=2h+v, n=c) both map to X(kbase + 2h + v, c),
    //   so the SAME register pair feeds SRC0 and SRC1 (symmetric Gram product).
    // (lane&7) folds cols 8..15 onto 0..7: a same-address LDS broadcast whose
    // products only land in unused rows/cols (>=6) of C.
    const float* pbase = Xrow + ((lane >> 4) << 4) + (lane & 7);

    v8f acc = {};
    #pragma unroll 8
    for (int g = 0; g < 64; ++g) {
        v2f av;
        av.x = pbase[g * 32];
        av.y = pbase[g * 32 + 8];
        acc = __builtin_amdgcn_wmma_f32_16x16x4_f32(
            /*neg_a=*/false, av, /*neg_b=*/false, av,
            /*c_mod=*/(short)0, acc, /*reuse_a=*/false, /*reuse_b=*/false);
    }

    // ---------------- Phase 3: extract 6x6 corner (rows=VGPR, cols=lane) -----
    if (lane < 6) {
        #pragma unroll
        for (int r = 0; r < 6; ++r)
            sol[r * 6 + lane] = acc[r];
    }

    // ---------------- Phase 4: LM-damped 5x5 solve + update (lane 0) ---------
    if (lane == 0) {
        // M[p][q] = -Hessian[p][q] (p,q<5),  M[p][5] = grad[p]
        // Solve (M with diag*(1+lambda)) * dtheta = grad  ==  -Bmat*dtheta = grad
        float A[5][6];
        #pragma unroll
        for (int p = 0; p < 5; ++p) {
            #pragma unroll
            for (int q = 0; q < 6; ++q)
                A[p][q] = sol[p * 6 + q];
            A[p][p] *= (1.0f + LAMBDA_);
        }
        // Gaussian elimination (SPD Gram + positive damping: no pivoting)
        #pragma unroll
        for (int col = 0; col < 4; ++col) {
            const float inv = 1.0f / A[col][col];
            #pragma unroll
            for (int row = col + 1; row < 5; ++row) {
                const float f = A[row][col] * inv;
                #pragma unroll
                for (int q = col; q < 6; ++q)
                    A[row][q] -= f * A[col][q];
            }
        }
        float x[5];
        #pragma unroll
        for (int row = 4; row >= 0; --row) {
            float v = A[row][5];
            #pragma unroll
            for (int q = row + 1; q < 5; ++q)
                v -= A[row][q] * x[q];
            x[row] = v / A[row][row];
        }
        const float* curB = cur + (size_t)b * 5;
        float*       outB = out + (size_t)b * 5;
        #pragma unroll
        for (int p = 0; p < 5; ++p) {
            const float c0 = curB[p];
            float d = x[p];
            if (__builtin_isnan(d)) d = -0.1f * c0;
            float v = c0 + d;
            const float mn = prm[p * 2 + 0];
            const float mx = prm[p * 2 + 1];
            v = fminf(fmaxf(v, mn), mx);
            outB[p] = v;
        }
    }
}

extern "C" void kernel_launch(void* const* d_in, const int* in_sizes, int n_in,
                              void* d_out, int out_size, void* d_ws, size_t ws_size,
                              hipStream_t stream) {
    const float* smp = (const float*)d_in[0];
    const float* mu  = (const float*)d_in[1];
    const float* jac = (const float*)d_in[2];
    const float* cur = (const float*)d_in[3];
    const float* prm = (const float*)d_in[4];
    float* out = (float*)d_out;

    const int B = in_sizes[0] / 256;   // smp is [B,16,16]
    if (B <= 0) return;
    const int blocks = (B + WAVES_PER_BLOCK - 1) / WAVES_PER_BLOCK;
    hipLaunchKernelGGL(lm_mle_wmma_kernel, dim3(blocks), dim3(WAVES_PER_BLOCK * 32),
                       0, stream, smp, mu, jac, cur, prm, out, B);
}